// GraphAttentionWildfire_28278064676999
// MI455X (gfx1250) — compile-verified
//
#include <hip/hip_runtime.h>
#include <math.h>

#define SLOPE 0.2f

typedef __attribute__((ext_vector_type(2))) float v2f;
typedef __attribute__((ext_vector_type(8))) float v8f;

// ---------------- utility ----------------

__global__ void fill_zero_k(float* p, int n) {
  int i = blockIdx.x * blockDim.x + threadIdx.x;
  if (i < n) p[i] = 0.0f;
}

__device__ __forceinline__ unsigned f2ord(float f) {
  unsigned u = __float_as_uint(f);
  return (u & 0x80000000u) ? ~u : (u | 0x80000000u);
}
__device__ __forceinline__ float ord2f(unsigned o) {
  return (o & 0x80000000u) ? __uint_as_float(o & 0x7fffffffu) : __uint_as_float(~o);
}
__device__ __forceinline__ float lrelu(float x) { return x > 0.0f ? x : SLOPE * x; }

// ---------------- fp32 WMMA GEMM: C[M,NC] = A[M,K] * B[K,NC] ----------------
// One wave32 computes one 16x16 tile via V_WMMA_F32_16X16X4_F32.
// A 16x4 layout: lane = 16*half + m; VGPR0/1 hold K = 2*half, 2*half+1 for row m.
// B  4x16 layout (mirrored): lane = 16*half + n; VGPR0/1 hold K = 2*half, 2*half+1 for col n.
// C/D: VGPR r holds C[M = 8*half + r][N = lane&15].

__global__ void gemm_wmma_k(const float* __restrict__ A, const float* __restrict__ B,
                            float* __restrict__ C, int M, int K, int NC) {
  int gid  = blockIdx.x * blockDim.x + threadIdx.x;
  int wave = gid >> 5;
  int lane = threadIdx.x & 31;
  int tilesN = NC >> 4;
  int tm = wave / tilesN;
  int tn = wave - tm * tilesN;
  if (tm * 16 >= M) return;              // wave-uniform: EXEC stays all-1s for WMMA

  int mr   = lane & 15;
  int half = lane >> 4;
  const float* arow = A + (size_t)(tm * 16 + mr) * K;
  const float* bcol = B + (tn * 16 + mr);

  v8f acc = {};
  for (int k = 0; k < K; k += 4) {
    int ka = k + half * 2;
    v2f a, b;
    a[0] = arow[ka];
    a[1] = arow[ka + 1];
    b[0] = bcol[(size_t)ka * NC];
    b[1] = bcol[(size_t)(ka + 1) * NC];
    acc = __builtin_amdgcn_wmma_f32_16x16x4_f32(false, a, false, b, (short)0, acc,
                                                false, false);
  }
  float* crow = C + (size_t)(tm * 16 + half * 8) * NC + tn * 16 + mr;
#pragma unroll
  for (int r = 0; r < 8; ++r) crow[(size_t)r * NC] = acc[r];
}

// ---------------- attention coefficients: a_s/a_d [N,heads] ----------------

__global__ void attn_coeff_k(const float* __restrict__ h, const float* __restrict__ attS,
                             const float* __restrict__ attD, float* __restrict__ a_s,
                             float* __restrict__ a_d, int n, int heads) {
  int tid = blockIdx.x * blockDim.x + threadIdx.x;
  if (tid >= n * heads) return;
  int node = tid / heads, hd = tid - node * heads;
  const float* hp = h + ((size_t)node * heads + hd) * 64;
  const float* sp = attS + hd * 64;
  const float* dp = attD + hd * 64;
  float s = 0.0f, d = 0.0f;
#pragma unroll 8
  for (int c = 0; c < 64; ++c) {
    float v = hp[c];
    s += v * sp[c];
    d += v * dp[c];
  }
  a_s[tid] = s;
  a_d[tid] = d;
}

// ---------------- edge passes (self-loops appended implicitly) ----------------

__device__ __forceinline__ void edge_nodes(const int* __restrict__ srcA,
                                           const int* __restrict__ dstA, int e, int E0,
                                           int& s, int& d) {
  if (e < E0) { s = srcA[e]; d = dstA[e]; }
  else        { s = d = e - E0; }
}

__global__ void edge_max_k(const int* __restrict__ srcA, const int* __restrict__ dstA,
                           const float* __restrict__ a_s, const float* __restrict__ a_d,
                           unsigned* __restrict__ emax, int E0, int n, int heads) {
  int tid = blockIdx.x * blockDim.x + threadIdx.x;
  int tot = (E0 + n) * heads;
  if (tid >= tot) return;
  int e = tid / heads, hd = tid - e * heads;
  int s, d; edge_nodes(srcA, dstA, e, E0, s, d);
  float ev = lrelu(a_s[s * heads + hd] + a_d[d * heads + hd]);
  atomicMax(emax + d * heads + hd, f2ord(ev));
}

__global__ void edge_sum_k(const int* __restrict__ srcA, const int* __restrict__ dstA,
                           const float* __restrict__ a_s, const float* __restrict__ a_d,
                           const unsigned* __restrict__ emax, float* __restrict__ denom,
                           int E0, int n, int heads) {
  int tid = blockIdx.x * blockDim.x + threadIdx.x;
  int tot = (E0 + n) * heads;
  if (tid >= tot) return;
  int e = tid / heads, hd = tid - e * heads;
  int s, d; edge_nodes(srcA, dstA, e, E0, s, d);
  int dh = d * heads + hd;
  float ev = lrelu(a_s[s * heads + hd] + a_d[dh]);
  atomicAdd(denom + dh, __expf(ev - ord2f(emax[dh])));
}

__global__ void edge_agg_k(const int* __restrict__ srcA, const int* __restrict__ dstA,
                           const float* __restrict__ a_s, const float* __restrict__ a_d,
                           const unsigned* __restrict__ emax, const float* __restrict__ denom,
                           const float* __restrict__ h, float* __restrict__ agg,
                           int E0, int n, int heads) {
  int tid = blockIdx.x * blockDim.x + threadIdx.x;
  int tot = (E0 + n) * heads;
  if (tid >= tot) return;
  int e = tid / heads, hd = tid - e * heads;
  int s, d; edge_nodes(srcA, dstA, e, E0, s, d);
  int dh = d * heads + hd;
  float ev = lrelu(a_s[s * heads + hd] + a_d[dh]);
  float alpha = __expf(ev - ord2f(emax[dh])) / denom[dh];
  const float* hp = h + ((size_t)s * heads + hd) * 64;
  float* op = agg + ((size_t)d * heads + hd) * 64;
#pragma unroll 8
  for (int c = 0; c < 64; ++c) atomicAdd(op + c, hp[c] * alpha);
}

// ---------------- bias + ELU epilogue ----------------

__global__ void finalize_k(const float* __restrict__ agg, const float* __restrict__ bias,
                           float* __restrict__ act, int n, int HC) {
  int tid = blockIdx.x * blockDim.x + threadIdx.x;
  if (tid >= n * HC) return;
  int c = tid % HC;
  float v = agg[tid] + bias[c];
  act[tid] = v > 0.0f ? v : (__expf(v) - 1.0f);
}

// ---------------- output MLP: relu(h@w1+b1)@w2+b2, weights staged in LDS ----------------

__global__ void mlp_k(const float* __restrict__ act, const float* __restrict__ w1,
                      const float* __restrict__ b1, const float* __restrict__ w2,
                      const float* __restrict__ b2, float* __restrict__ out, int n) {
  __shared__ float sw1[64 * 32];
  __shared__ float sb1[32];
  __shared__ float sw2[32];
  __shared__ float sb2;
  for (int i = threadIdx.x; i < 64 * 32; i += blockDim.x) sw1[i] = w1[i];
  if (threadIdx.x < 32) { sb1[threadIdx.x] = b1[threadIdx.x]; sw2[threadIdx.x] = w2[threadIdx.x]; }
  if (threadIdx.x == 0) sb2 = b2[0];
  __syncthreads();

  int node = blockIdx.x * blockDim.x + threadIdx.x;
  if (node >= n) return;
  const float* xp = act + (size_t)node * 64;
  float x[64];
#pragma unroll
  for (int i = 0; i < 64; ++i) x[i] = xp[i];
  float o = sb2;
#pragma unroll 4
  for (int j = 0; j < 32; ++j) {
    float a = sb1[j];
#pragma unroll 8
    for (int i = 0; i < 64; ++i) a += x[i] * sw1[i * 32 + j];
    a = a > 0.0f ? a : 0.0f;
    o += a * sw2[j];
  }
  out[node] = o;
}

// ---------------- host-side orchestration ----------------

extern "C" void kernel_launch(void* const* d_in, const int* in_sizes, int n_in,
                              void* d_out, int out_size, void* d_ws, size_t ws_size,
                              hipStream_t stream) {
  const float* x    = (const float*)d_in[0];
  const int*   edge = (const int*)d_in[1];
  const float* lin0 = (const float*)d_in[2];
  const float* as0  = (const float*)d_in[3];
  const float* ad0  = (const float*)d_in[4];
  const float* bi0  = (const float*)d_in[5];
  const float* lin1 = (const float*)d_in[6];
  const float* as1  = (const float*)d_in[7];
  const float* ad1  = (const float*)d_in[8];
  const float* bi1  = (const float*)d_in[9];
  const float* lin2 = (const float*)d_in[10];
  const float* as2  = (const float*)d_in[11];
  const float* ad2  = (const float*)d_in[12];
  const float* bi2  = (const float*)d_in[13];
  const float* w1   = (const float*)d_in[14];
  const float* mb1  = (const float*)d_in[15];
  const float* w2   = (const float*)d_in[16];
  const float* mb2  = (const float*)d_in[17];

  const int N  = in_sizes[0] / 128;   // 20000 (multiple of 16)
  const int E  = in_sizes[1] / 2;     // 320000
  const int ET = E + N;               // with self-loops
  const int* srcA = edge;
  const int* dstA = edge + E;

  // workspace layout (floats): h | agg | act | a_s | a_d | emax(u32) | denom
  float* W     = (float*)d_ws;
  float* h     = W;
  float* agg   = h   + (size_t)N * 256;
  float* act   = agg + (size_t)N * 256;
  float* a_s   = act + (size_t)N * 256;
  float* a_d   = a_s + (size_t)N * 4;
  unsigned* em = (unsigned*)(a_d + (size_t)N * 4);
  float* denom = (float*)(em + (size_t)N * 4);

  auto nb = [](int n) { return (n + 255) / 256; };

  auto layer = [&](const float* Ain, int K, const float* lin, int heads,
                   const float* attS, const float* attD, const float* bias) {
    int HC = heads * 64;
    int waves = (N / 16) * (HC / 16);
    gemm_wmma_k<<<nb(waves * 32), 256, 0, stream>>>(Ain, lin, h, N, K, HC);
    attn_coeff_k<<<nb(N * heads), 256, 0, stream>>>(h, attS, attD, a_s, a_d, N, heads);
    fill_zero_k<<<nb(N * HC), 256, 0, stream>>>(agg, N * HC);
    fill_zero_k<<<nb(N * heads), 256, 0, stream>>>((float*)em, N * heads);   // 0u < ord(any float)
    fill_zero_k<<<nb(N * heads), 256, 0, stream>>>(denom, N * heads);
    edge_max_k<<<nb(ET * heads), 256, 0, stream>>>(srcA, dstA, a_s, a_d, em, E, N, heads);
    edge_sum_k<<<nb(ET * heads), 256, 0, stream>>>(srcA, dstA, a_s, a_d, em, denom, E, N, heads);
    edge_agg_k<<<nb(ET * heads), 256, 0, stream>>>(srcA, dstA, a_s, a_d, em, denom, h, agg, E, N, heads);
    finalize_k<<<nb(N * HC), 256, 0, stream>>>(agg, bias, act, N, HC);
  };

  layer(x,   128, lin0, 4, as0, ad0, bi0);   // IN=128 -> [N, 256]
  layer(act, 256, lin1, 4, as1, ad1, bi1);   // 256   -> [N, 256]
  layer(act, 256, lin2, 1, as2, ad2, bi2);   // 256   -> [N, 64] (1 head, mean == identity)

  mlp_k<<<nb(N), 256, 0, stream>>>(act, w1, mb1, w2, mb2, (float*)d_out, N);
}